// QuantumSelfAttention_65481071405315
// MI455X (gfx1250) — compile-verified
//
#include <hip/hip_runtime.h>
#include <math.h>

#define QSA_N    10          // EMBED_DIM
#define ROWS     128         // rows (tokens) per block
#define THREADS  128         // one thread per row
#define TILE_FLOATS (ROWS * QSA_N)     // 1280 floats
#define TILE_BYTES  (TILE_FLOATS * 4)  // 5120 bytes
#define TILE_B128   (TILE_BYTES / 16)  // 320 chunks: 2 full rounds + 64-lane round

// Pointee type the async builtins expect (from clang diagnostic): int4
typedef int v4i __attribute__((vector_size(4 * sizeof(int))));
typedef __attribute__((address_space(1))) v4i g_v4i;   // global int4
typedef __attribute__((address_space(3))) v4i l_v4i;   // LDS int4

#if defined(__AMDGCN__) && defined(__has_builtin)
#if __has_builtin(__builtin_amdgcn_global_load_async_to_lds_b128) && \
    __has_builtin(__builtin_amdgcn_s_wait_asynccnt)
#define QSA_ASYNC_LOAD 1
#endif
#if __has_builtin(__builtin_amdgcn_global_store_async_from_lds_b128) && \
    __has_builtin(__builtin_amdgcn_s_wait_asynccnt)
#define QSA_ASYNC_STORE 1
#endif
#endif

// Closed form of the reference circuit (CRY ladder; derived & hand-verified):
//   Z_0 = cos(x0+r0); q = (1-Z_0)/2
//   Z_k = c_k*(1 - 2*q*S_{k-1}),  c_k = cos(x_k+r_k),  S_i = (1-cos(e_i))/2
//   q  <- (1-c_k)/2 + q*S_{k-1}*c_k

// Main kernel: handles full 128-row tiles only (no guards, straight-line).
__global__ __launch_bounds__(THREADS) void qsa_expvals_kernel(
    const float* __restrict__ x,    // (T, N)
    const float* __restrict__ rot,  // (N,)
    const float* __restrict__ ent,  // (N-1,)
    float* __restrict__ out)        // (T, N)
{
    __shared__ float tile[TILE_FLOATS];
    __shared__ float sP[2 * QSA_N];   // [0..N-1]=rot, [N..2N-2]=S_i

    const int tid = threadIdx.x;
    const int t0  = (int)blockIdx.x * ROWS;

    const float* src = x   + t0 * QSA_N;
    float*       dst = out + t0 * QSA_N;

    // ---- stage input tile into LDS: gfx1250 async DMA, fully unrolled ----
#if QSA_ASYNC_LOAD
    {
        const char* gb = (const char*)src;
        char*       lb = (char*)tile;
        const int b0 = tid * 16;
        __builtin_amdgcn_global_load_async_to_lds_b128(
            (g_v4i*)(gb + b0), (l_v4i*)(lb + b0), 0, 0);
        __builtin_amdgcn_global_load_async_to_lds_b128(
            (g_v4i*)(gb + b0 + THREADS * 16), (l_v4i*)(lb + b0 + THREADS * 16), 0, 0);
        if (tid < TILE_B128 - 2 * THREADS)   // last 64 chunks
            __builtin_amdgcn_global_load_async_to_lds_b128(
                (g_v4i*)(gb + b0 + 2 * THREADS * 16), (l_v4i*)(lb + b0 + 2 * THREADS * 16), 0, 0);
        __builtin_amdgcn_s_wait_asynccnt(0);
    }
#else
    for (int i = tid; i < TILE_FLOATS; i += THREADS) tile[i] = src[i];
#endif

    if (tid < QSA_N)     sP[tid]         = rot[tid];
    if (tid < QSA_N - 1) sP[QSA_N + tid] = 0.5f * (1.0f - __cosf(ent[tid]));
    __syncthreads();

    // ---- per-row closed-form recurrence (each thread owns one row) ----
    {
        float* row = &tile[tid * QSA_N];   // stride 10 -> conflict-free in wave32
        float o[QSA_N];

        float c0 = __cosf(row[0] + sP[0]);
        o[0]     = c0;
        float q  = 0.5f * (1.0f - c0);     // P(qubit0 == 1)

#pragma unroll
        for (int k = 1; k < QSA_N; ++k) {
            float ck = __cosf(row[k] + sP[k]);
            float S  = sP[QSA_N + k - 1];
            float u  = q * S * ck;
            o[k] = ck - 2.0f * u;                  // ck*(1 - 2*q*S)
            q    = 0.5f - 0.5f * ck + u;           // next P(qubit k == 1)
        }
#pragma unroll
        for (int k = 0; k < QSA_N; ++k) row[k] = o[k];   // only this thread touched row
    }
    __syncthreads();

    // ---- stream tile back out: gfx1250 async LDS->global, fully unrolled ----
#if QSA_ASYNC_STORE
    {
        char* gb = (char*)dst;
        char* lb = (char*)tile;
        const int b0 = tid * 16;
        __builtin_amdgcn_global_store_async_from_lds_b128(
            (g_v4i*)(gb + b0), (l_v4i*)(lb + b0), 0, 0);
        __builtin_amdgcn_global_store_async_from_lds_b128(
            (g_v4i*)(gb + b0 + THREADS * 16), (l_v4i*)(lb + b0 + THREADS * 16), 0, 0);
        if (tid < TILE_B128 - 2 * THREADS)
            __builtin_amdgcn_global_store_async_from_lds_b128(
                (g_v4i*)(gb + b0 + 2 * THREADS * 16), (l_v4i*)(lb + b0 + 2 * THREADS * 16), 0, 0);
        __builtin_amdgcn_s_wait_asynccnt(0);
    }
#else
    for (int i = tid; i < TILE_FLOATS; i += THREADS) dst[i] = tile[i];
#endif
}

// Tail kernel: one thread per leftover row, direct global access (cold path;
// never launched for the reference shapes where T % ROWS == 0).
__global__ __launch_bounds__(THREADS) void qsa_tail_kernel(
    const float* __restrict__ x, const float* __restrict__ rot,
    const float* __restrict__ ent, float* __restrict__ out,
    int tStart, int T)
{
    const int t = tStart + (int)blockIdx.x * THREADS + (int)threadIdx.x;
    if (t >= T) return;
    const float* row = x   + t * QSA_N;
    float*       o   = out + t * QSA_N;

    float c0 = __cosf(row[0] + rot[0]);
    o[0]     = c0;
    float q  = 0.5f * (1.0f - c0);
#pragma unroll
    for (int k = 1; k < QSA_N; ++k) {
        float ck = __cosf(row[k] + rot[k]);
        float S  = 0.5f * (1.0f - __cosf(ent[k - 1]));
        float u  = q * S * ck;
        o[k] = ck - 2.0f * u;
        q    = 0.5f - 0.5f * ck + u;
    }
}

extern "C" void kernel_launch(void* const* d_in, const int* in_sizes, int n_in,
                              void* d_out, int out_size, void* d_ws, size_t ws_size,
                              hipStream_t stream) {
    const float* x   = (const float*)d_in[0];   // (16,1024,10) f32
    const float* rot = (const float*)d_in[1];   // (10,)        f32
    const float* ent = (const float*)d_in[2];   // (9,)         f32
    float*       out = (float*)d_out;           // (16,1024,10) f32

    const int n = in_sizes[1];                  // EMBED_DIM == QSA_N == 10
    const int T = in_sizes[0] / n;              // 16384 rows
    const int fullBlocks = T / ROWS;            // 128 for reference shapes
    const int tail       = T - fullBlocks * ROWS;

    if (fullBlocks > 0)
        qsa_expvals_kernel<<<fullBlocks, THREADS, 0, stream>>>(x, rot, ent, out);
    if (tail > 0)
        qsa_tail_kernel<<<(tail + THREADS - 1) / THREADS, THREADS, 0, stream>>>(
            x, rot, ent, out, fullBlocks * ROWS, T);

    (void)n_in; (void)out_size; (void)d_ws; (void)ws_size;
}